// EARLIEST_81681688035816
// MI455X (gfx1250) — compile-verified
//
#include <hip/hip_runtime.h>

typedef __attribute__((ext_vector_type(2))) float v2f;
typedef __attribute__((ext_vector_type(8))) float v8f;

constexpr int TT = 48;    // time
constexpr int BB = 64;    // batch
constexpr int VV = 128;   // features
constexpr int HH = 256;   // hidden
constexpr int CC = 10;    // classes
constexpr int GH = 768;   // 3*HH
constexpr int KC = HH / 4;        // 64 K-chunks of 4 for the recurrent GEMM
constexpr int WB_ELEMS = 16 * KC * 3 * 64;   // 196608 = GH*HH

__device__ __forceinline__ float sigm(float x) { return 1.0f / (1.0f + expf(-x)); }

__device__ __forceinline__ v8f wmma_f32(v2f a, v2f b, v8f c) {
  // V_WMMA_F32_16X16X4_F32: full-fp32 matrix core op (precision-faithful to reference)
  return __builtin_amdgcn_wmma_f32_16x16x4_f32(false, a, false, b, (short)0, c,
                                               false, false);
}

// ---------------------------------------------------------------------------
// Kernel 0: weight repack.
//  - W_ih (768x128) -> WihT (128x768) for the one-shot GI GEMM.
//  - W_hh (768x256) -> WB, fragment-native order: for each (col-tile ct,
//    K-chunk kc, gate) the exact 32-lane x 2-dword WMMA B fragment is stored
//    contiguously (256B), so the scan kernel streams B with single b64 loads.
//    WB[((ct*64+kc)*3+gate)*64 + lane*2 + sub] =
//        W_hh[(gate*256 + ct*16 + lane%16) * 256 + (kc*4 + 2*(lane/16) + sub)]
// ---------------------------------------------------------------------------
__global__ void repack_w(const float* __restrict__ W_ih,
                         const float* __restrict__ W_hh,
                         float* __restrict__ WihT,
                         float* __restrict__ WB) {
  int tid = blockIdx.x * blockDim.x + threadIdx.x;
  const int n_ih = GH * VV;   // 98304
  if (tid < n_ih) {
    int j = tid / VV, k = tid % VV;
    WihT[k * GH + j] = W_ih[j * VV + k];
  } else if (tid < n_ih + WB_ELEMS) {
    int o = tid - n_ih;
    int sub = o & 1;
    int l = (o >> 1) & 31;
    int r1 = o >> 6;           // (ct*64+kc)*3 + gate
    int gate = r1 % 3;
    int r2 = r1 / 3;           // ct*64 + kc
    int kc = r2 & 63;
    int ct = r2 >> 6;
    int row = gate * HH + ct * 16 + (l & 15);
    int k = kc * 4 + 2 * (l >> 4) + sub;
    WB[o] = W_hh[row * HH + k];
  }
}

// ---------------------------------------------------------------------------
// Kernel 1: GI[t,b,:] = X[t,b,:] @ W_ih^T + b_ih.  M=T*B=3072, N=768, K=128.
// One wave per 16x16 output tile, fp32 WMMA. One-shot (0.6 GFLOP).
// ---------------------------------------------------------------------------
__global__ void gi_kernel(const float* __restrict__ X,
                          const float* __restrict__ WihT,
                          const float* __restrict__ b_ih,
                          float* __restrict__ GI) {
  const int lane = threadIdx.x & 31;
  const int wave = (blockIdx.x * blockDim.x + threadIdx.x) >> 5;
  const int n_ct = GH / 16;                 // 48 col tiles
  const int rt = wave / n_ct;               // 0..191 row tiles (M=3072)
  const int ct = wave % n_ct;
  if (rt >= (TT * BB) / 16) return;         // wave-uniform guard (EXEC stays full)
  const int hi = lane >> 4;                 // half-wave
  const int lm = lane & 15;
  const int m0 = rt * 16, n0 = ct * 16;

  v8f acc = {};
#pragma unroll 4
  for (int k0 = 0; k0 < VV; k0 += 4) {
    const int ka = k0 + 2 * hi;
    v2f a, b;
    // A 16x4: lane -> M=lm, VGPR g + half-wave -> K = k0 + g + 2*hi
    a.x = X[(m0 + lm) * VV + ka + 0];
    a.y = X[(m0 + lm) * VV + ka + 1];
    // B 4x16: VGPR g + half-wave -> K, lane -> N
    b.x = WihT[(size_t)(ka + 0) * GH + n0 + lm];
    b.y = WihT[(size_t)(ka + 1) * GH + n0 + lm];
    acc = wmma_f32(a, b, acc);
  }
  const float bj = b_ih[n0 + lm];
#pragma unroll
  for (int g = 0; g < 8; ++g) {
    const int m = m0 + g + 8 * hi;          // C/D: VGPR g -> rows g and g+8
    GI[(size_t)m * GH + n0 + lm] = acc[g] + bj;
  }
}

// ---------------------------------------------------------------------------
// Kernel 2: the serial scan. One workgroup = 32 wave32 waves on one WGP.
// h kept in LDS twice: hT[k][b] (b-major, for gate math + head) and HA
// (WMMA-A fragment order, so A loads are single ds_load_b64). Per GRU step a
// wave computes 2 row-tiles x 1 col-tile x 3 gates of gh = h @ W_hh^T with
// fp32 WMMA, streaming B fragments from the repacked WB with b64 loads.
// ---------------------------------------------------------------------------
__global__ __launch_bounds__(1024, 1)
void gru_scan_kernel(const float* __restrict__ GI,
                     const float* __restrict__ WB,
                     const float* __restrict__ b_hh,
                     const float* __restrict__ U,
                     const float* __restrict__ W_out,
                     const float* __restrict__ b_out,
                     const float* __restrict__ W_c,
                     const float* __restrict__ b_c,
                     float* __restrict__ out) {
  __shared__ float hT[HH * BB];        // 64 KB: hT[k*BB + b]
  __shared__ float HA[4 * KC * 64];    // 64 KB: A fragments, [rt][kc][lane*2+sub]
  __shared__ float logitsS[BB * CC];   // 2.5 KB
  __shared__ float predsS[BB * CC];    // 2.5 KB
  __shared__ float haltS[BB];
  __shared__ float aS[BB];

  const int tid  = threadIdx.x;
  const int lane = tid & 31;
  const int wave = tid >> 5;           // 0..31
  const int hi   = lane >> 4;
  const int lm   = lane & 15;

  // h0 = 0 (both layouts), preds0 = 0, halt0 = -1
  for (int i = tid; i < HH * BB; i += 1024) { hT[i] = 0.0f; HA[i] = 0.0f; }
  if (tid < BB * CC) predsS[tid] = 0.0f;
  if (tid < BB) haltS[tid] = -1.0f;
  __syncthreads();

  const int ct  = wave & 15;           // column tile of H (16 tiles)
  const int rt0 = (wave >> 4) * 2;     // row tiles {0,1} or {2,3}
  const int rt1 = rt0 + 1;
  const int n0  = ct * 16;
  const int j   = n0 + lm;             // this lane's output column within H

  const float bhr = b_hh[j];
  const float bhz = b_hh[HH + j];
  const float bhn = b_hh[2 * HH + j];

  // per-thread write bases into the HA fragment layout for column j
  const int kcj  = j >> 2;
  const int frj  = 16 * ((j >> 1) & 1) * 2 + (j & 1);   // lane_r offset part
  const int haw0 = (rt0 * KC + kcj) * 64 + frj;
  const int haw1 = (rt1 * KC + kcj) * 64 + frj;

  const float* wb  = WB + (size_t)ct * (KC * 3 * 64) + lane * 2;
  const float* ha0 = HA + rt0 * (KC * 64) + lane * 2;
  const float* ha1 = HA + rt1 * (KC * 64) + lane * 2;

  for (int t = 1; t < TT; ++t) {
    for (int s = 0; s < t; ++s) {
      v8f ar0 = {}, az0 = {}, an0 = {};
      v8f ar1 = {}, az1 = {}, an1 = {};
#pragma unroll 4
      for (int kc = 0; kc < KC; ++kc) {
        const v2f A0 = *(const v2f*)(ha0 + kc * 64);
        const v2f A1 = *(const v2f*)(ha1 + kc * 64);
        const float* wp = wb + kc * 192;
        const v2f Br = *(const v2f*)(wp);
        const v2f Bz = *(const v2f*)(wp + 64);
        const v2f Bq = *(const v2f*)(wp + 128);
        ar0 = wmma_f32(A0, Br, ar0);
        az0 = wmma_f32(A0, Bz, az0);
        an0 = wmma_f32(A0, Bq, an0);
        ar1 = wmma_f32(A1, Br, ar1);
        az1 = wmma_f32(A1, Bz, az1);
        an1 = wmma_f32(A1, Bq, an1);
      }
      // GRU gate math directly on this wave's accumulators.
      const float* gi = GI + (size_t)s * BB * GH;
      float hn0[8], hn1[8];
#pragma unroll
      for (int g = 0; g < 8; ++g) {
        {
          const int b = rt0 * 16 + g + 8 * hi;
          const float gr = gi[(size_t)b * GH + j];
          const float gz = gi[(size_t)b * GH + HH + j];
          const float gn = gi[(size_t)b * GH + 2 * HH + j];
          const float hold = hT[j * BB + b];
          const float r = sigm(gr + ar0[g] + bhr);
          const float z = sigm(gz + az0[g] + bhz);
          const float n = tanhf(gn + r * (an0[g] + bhn));
          hn0[g] = (1.0f - z) * n + z * hold;
        }
        {
          const int b = rt1 * 16 + g + 8 * hi;
          const float gr = gi[(size_t)b * GH + j];
          const float gz = gi[(size_t)b * GH + HH + j];
          const float gn = gi[(size_t)b * GH + 2 * HH + j];
          const float hold = hT[j * BB + b];
          const float r = sigm(gr + ar1[g] + bhr);
          const float z = sigm(gz + az1[g] + bhz);
          const float n = tanhf(gn + r * (an1[g] + bhn));
          hn1[g] = (1.0f - z) * n + z * hold;
        }
      }
      __syncthreads();   // all waves done reading old h
#pragma unroll
      for (int g = 0; g < 8; ++g) {
        const int bl = g + 8 * hi;         // b%16 for this element
        hT[j * BB + rt0 * 16 + bl] = hn0[g];
        hT[j * BB + rt1 * 16 + bl] = hn1[g];
        HA[haw0 + bl * 2] = hn0[g];        // fragment layout copy
        HA[haw1 + bl * 2] = hn1[g];
      }
      __syncthreads();   // new h visible to all waves
    }

    // ---- halting head at outer step t ----
    if (tid < BB * CC) {               // logits_t = h @ W_out^T + b_out
      const int b = tid / CC, c = tid % CC;
      float acc = b_out[c];
      for (int jj = 0; jj < HH; ++jj) acc += hT[jj * BB + b] * W_out[c * HH + jj];
      logitsS[tid] = acc;
    }
    if (tid >= 640 && tid < 640 + BB) {  // p = sigmoid([h, t] @ W_c^T + b_c); a = U<p
      const int b = tid - 640;
      float acc = b_c[0] + (float)t * W_c[HH];
      for (int jj = 0; jj < HH; ++jj) acc += hT[jj * BB + b] * W_c[jj];
      const float p = sigm(acc);
      aS[b] = (U[(size_t)t * BB + b] < p) ? 1.0f : 0.0f;
    }
    __syncthreads();
    if (tid < BB * CC) {
      const int b = tid / CC;
      if (aS[b] == 1.0f && predsS[tid] == 0.0f) predsS[tid] = logitsS[tid];
    }
    if (tid >= 640 && tid < 640 + BB) {
      const int b = tid - 640;
      if (haltS[b] == -1.0f && aS[b] == 1.0f) haltS[b] = (float)t;
    }
    __syncthreads();
  }

  // ---- finalize: logits = where(preds==0, logits_{T-1}, preds); halt_mean ----
  if (tid < BB * CC) {
    const float pv = predsS[tid];
    out[tid] = (pv == 0.0f) ? logitsS[tid] : pv;
  }
  if (tid == 0) {
    float ssum = 0.0f;
    for (int b = 0; b < BB; ++b) {
      float hb = haltS[b];
      if (hb == -1.0f) hb = (float)(TT - 1);
      ssum += 1.0f + hb;
    }
    out[BB * CC] = ssum / (float)BB / (float)(TT + 1);
  }
}

// ---------------------------------------------------------------------------
extern "C" void kernel_launch(void* const* d_in, const int* in_sizes, int n_in,
                              void* d_out, int out_size, void* d_ws, size_t ws_size,
                              hipStream_t stream) {
  const float* X     = (const float*)d_in[0];
  const float* U     = (const float*)d_in[1];
  const float* W_ih  = (const float*)d_in[2];
  const float* W_hh  = (const float*)d_in[3];
  const float* b_ih  = (const float*)d_in[4];
  const float* b_hh  = (const float*)d_in[5];
  const float* W_out = (const float*)d_in[6];
  const float* b_out = (const float*)d_in[7];
  const float* W_c   = (const float*)d_in[8];
  const float* b_c   = (const float*)d_in[9];
  // W_b / b_b (d_in[10..11]) do not affect the returned outputs.

  float* out  = (float*)d_out;            // 641 floats: logits(640) + halt_mean
  float* ws   = (float*)d_ws;
  float* WihT = ws;                       // 128*768
  float* WB   = ws + (size_t)GH * VV;     // 196608 (fragment-packed W_hh)
  float* GI   = WB + WB_ELEMS;            // 48*64*768

  // 0) repack weights (98304 + 196608 elems)
  repack_w<<<1152, 256, 0, stream>>>(W_ih, W_hh, WihT, WB);
  // 1) GI precompute: 9216 tiles, 8 waves/block -> 1152 blocks
  gi_kernel<<<1152, 256, 0, stream>>>(X, WihT, b_ih, GI);
  // 2) serial adaptive-halting GRU scan: single WGP-resident workgroup
  gru_scan_kernel<<<1, 1024, 0, stream>>>(GI, WB, b_hh, U, W_out, b_out,
                                          W_c, b_c, out);
}